// SS2D_19447611916707
// MI455X (gfx1250) — compile-verified
//
#include <hip/hip_runtime.h>
#include <math.h>

// ---------------------------------------------------------------------------
// SS2D (VMamba selective-scan 2D) for MI455X / gfx1250, fp32 end-to-end.
//  * All 4 GEMM stages use V_WMMA_F32_16X16X4_F32 (wave32, 16x16 tile/wave).
//  * Selective scan implemented as a 3-pass chunked linear scan (64 chunks of
//    64): per-chunk affine composition (P,Q), chunk-prefix, in-chunk replay.
//  * Fast-path transcendentals: v_rcp_f32 for SiLU, v_rsq_f32 for LayerNorm
//    (avoids the IEEE div/sqrt expansion chains seen in round-1 asm).
// ---------------------------------------------------------------------------

#define DM 96            // d_model
#define DI 192           // d_inner
#define NS 16            // d_state
#define RK 6             // dt_rank
#define KD 4             // scan directions
#define BB 2
#define HH 64
#define WW 64
#define LL (HH * WW)     // 4096
#define CC (RK + 2 * NS) // 38 rows of x_dbl
#define NCHUNK 64
#define CLEN (LL / NCHUNK) // 64
#define NCH (BB * KD * DI) // 1536 scan channels

typedef __attribute__((ext_vector_type(2))) float v2f;
typedef __attribute__((ext_vector_type(8))) float v8f;

__device__ __forceinline__ v8f wmma_f32_k4(v2f a, v2f b, v8f c) {
  // D(16x16,f32) = A(16x4,f32) * B(4x16,f32) + C
  return __builtin_amdgcn_wmma_f32_16x16x4_f32(false, a, false, b, (short)0, c,
                                               false, false);
}

__device__ __forceinline__ float silu_f(float x) {
  // x * sigmoid(x); v_rcp_f32 (~1 ulp) instead of IEEE division expansion
  return x * __builtin_amdgcn_rcpf(1.f + __expf(-x));
}
__device__ __forceinline__ float softplus_f(float x) {
  return (x > 20.f) ? x : log1pf(__expf(x));
}

// ---------------------------------------------------------------------------
// K1: xz = x @ in_proj_w^T   (M = B*L = 8192, N = 2*DI = 384, K = 96)
//     columns [0,DI)   -> xi   (conv input)
//     columns [DI,2DI) -> z = silu(.)
// one wave per 16x16 tile; K-loop of 24 v_wmma_f32_16x16x4_f32
// ---------------------------------------------------------------------------
__global__ void k_inproj(const float* __restrict__ x, const float* __restrict__ w,
                         float* __restrict__ xi, float* __restrict__ z) {
  const int wid = (blockIdx.x * blockDim.x + threadIdx.x) >> 5;
  const int lane = threadIdx.x & 31;
  const int TN = (2 * DI) / 16; // 24
  const int tm = wid / TN, tn = wid % TN;
  if (tm >= (BB * LL) / 16) return;
  const int m0 = tm * 16, n0 = tn * 16;
  const int r = lane & 15, hk = lane >> 4;
  const float* arow = x + (size_t)(m0 + r) * DM;
  const float* brow = w + (size_t)(n0 + r) * DM;
  v8f acc = {0.f, 0.f, 0.f, 0.f, 0.f, 0.f, 0.f, 0.f};
#pragma unroll 4
  for (int kk = 0; kk < DM; kk += 4) {
    const int kb = kk + 2 * hk;
    v2f a, b;
    a.x = arow[kb];     a.y = arow[kb + 1];
    b.x = brow[kb];     b.y = brow[kb + 1];
    acc = wmma_f32_k4(a, b, acc);
  }
#pragma unroll
  for (int i = 0; i < 8; ++i) {
    const int m = m0 + i + 8 * hk;
    const int n = n0 + r;
    const float v = acc[i];
    if (n < DI) xi[(size_t)m * DI + n] = v;
    else        z[(size_t)m * DI + (n - DI)] = silu_f(v);
  }
}

// ---------------------------------------------------------------------------
// K2: depthwise 3x3 conv (+bias, silu) on xi (B,H,W,DI), scattered directly
//     into the 4 directional sequences xs (B,K,DI,L).
// ---------------------------------------------------------------------------
__global__ void k_conv_silu_scatter(const float* __restrict__ xi,
                                    const float* __restrict__ cw,
                                    const float* __restrict__ cb,
                                    float* __restrict__ xs) {
  const int idx = blockIdx.x * blockDim.x + threadIdx.x;
  if (idx >= BB * DI * LL) return;
  const int w = idx % WW;
  const int h = (idx / WW) % HH;
  const int c = (idx / LL) % DI;
  const int b = idx / (DI * LL);
  float s = cb[c];
#pragma unroll
  for (int dh = -1; dh <= 1; ++dh) {
    const int h2 = h + dh;
    if (h2 < 0 || h2 >= HH) continue;
#pragma unroll
    for (int dw = -1; dw <= 1; ++dw) {
      const int w2 = w + dw;
      if (w2 < 0 || w2 >= WW) continue;
      s += xi[((size_t)(b * HH + h2) * WW + w2) * DI + c] *
           cw[c * 9 + (dh + 1) * 3 + (dw + 1)];
    }
  }
  const float v = silu_f(s);
  const int lhw = h * WW + w;
  const int lwh = w * HH + h;
  const size_t base = (size_t)b * KD * DI * LL + (size_t)c * LL;
  const size_t kstr = (size_t)DI * LL;
  xs[base + 0 * kstr + lhw] = v;                // HW order
  xs[base + 1 * kstr + lwh] = v;                // WH order
  xs[base + 2 * kstr + (LL - 1 - lhw)] = v;     // flipped HW
  xs[base + 3 * kstr + (LL - 1 - lwh)] = v;     // flipped WH
}

// ---------------------------------------------------------------------------
// K3: x_dbl[b,k] = x_proj_weight[k] (38x192) @ xs[b,k] (192xL)
//     M tiles = 3 (rows 38..47 masked), N tiles = 256, K steps = 48
// ---------------------------------------------------------------------------
__global__ void k_xdbl(const float* __restrict__ xs, const float* __restrict__ xpw,
                       float* __restrict__ dbl) {
  const int wid = (blockIdx.x * blockDim.x + threadIdx.x) >> 5;
  const int lane = threadIdx.x & 31;
  const int TM = 3, TN = LL / 16;
  const int tn = wid % TN;
  const int tm = (wid / TN) % TM;
  const int bk = wid / (TN * TM);
  if (bk >= BB * KD) return;
  const int k = bk % KD;
  const int r = lane & 15, hk = lane >> 4;
  const int l0 = tn * 16;
  const int row_a = tm * 16 + r;
  const float* A = xpw + (size_t)k * CC * DI;
  const float* Bm = xs + (size_t)bk * DI * LL;
  v8f acc = {0.f, 0.f, 0.f, 0.f, 0.f, 0.f, 0.f, 0.f};
  for (int kk = 0; kk < DI; kk += 4) {
    const int kb = kk + 2 * hk;
    v2f a, b;
    a.x = (row_a < CC) ? A[(size_t)row_a * DI + kb] : 0.f;
    a.y = (row_a < CC) ? A[(size_t)row_a * DI + kb + 1] : 0.f;
    b.x = Bm[(size_t)kb * LL + l0 + r];
    b.y = Bm[(size_t)(kb + 1) * LL + l0 + r];
    acc = wmma_f32_k4(a, b, acc);
  }
#pragma unroll
  for (int i = 0; i < 8; ++i) {
    const int cr = tm * 16 + i + 8 * hk;
    if (cr < CC) dbl[((size_t)bk * CC + cr) * LL + l0 + r] = acc[i];
  }
}

// ---------------------------------------------------------------------------
// K4: dt[b,k] = dt_projs_weight[k] (192x6) @ x_dbl[b,k][0:6,:] (6xL)
//     K padded 6 -> 8 (2 wmma steps with masked loads)
// ---------------------------------------------------------------------------
__global__ void k_dtproj(const float* __restrict__ dbl, const float* __restrict__ dtw,
                         float* __restrict__ dt) {
  const int wid = (blockIdx.x * blockDim.x + threadIdx.x) >> 5;
  const int lane = threadIdx.x & 31;
  const int TM = DI / 16, TN = LL / 16; // 12 x 256
  const int tn = wid % TN;
  const int tm = (wid / TN) % TM;
  const int bk = wid / (TN * TM);
  if (bk >= BB * KD) return;
  const int k = bk % KD;
  const int r = lane & 15, hk = lane >> 4;
  const int l0 = tn * 16;
  const int row_a = tm * 16 + r;
  const float* A = dtw + (size_t)k * DI * RK;
  const float* Bm = dbl + (size_t)bk * CC * LL; // rows 0..5 = raw dts
  v8f acc = {0.f, 0.f, 0.f, 0.f, 0.f, 0.f, 0.f, 0.f};
#pragma unroll
  for (int kk = 0; kk < 8; kk += 4) {
    const int kb = kk + 2 * hk;
    v2f a, b;
    a.x = (kb < RK)     ? A[(size_t)row_a * RK + kb] : 0.f;
    a.y = (kb + 1 < RK) ? A[(size_t)row_a * RK + kb + 1] : 0.f;
    b.x = (kb < RK)     ? Bm[(size_t)kb * LL + l0 + r] : 0.f;
    b.y = (kb + 1 < RK) ? Bm[(size_t)(kb + 1) * LL + l0 + r] : 0.f;
    acc = wmma_f32_k4(a, b, acc);
  }
#pragma unroll
  for (int i = 0; i < 8; ++i) {
    const int m = tm * 16 + i + 8 * hk;
    dt[((size_t)bk * DI + m) * LL + l0 + r] = acc[i];
  }
}

// ---------------------------------------------------------------------------
// Selective scan, chunked (h <- a*h + b is affine; 64-step chunk == (P,Q)).
// Pass 1: per (channel, chunk) compute P[16], Q[16].         98304 threads
// Pass 2: per (channel, state) prefix over 64 chunks -> h0.  24576 threads
// Pass 3: per (channel, chunk) replay from h0, emit y in-place over xs.
// ---------------------------------------------------------------------------
__global__ void k_scan_pass1(const float* __restrict__ xs, const float* __restrict__ dbl,
                             const float* __restrict__ dt,
                             const float* __restrict__ alog,
                             const float* __restrict__ dtb,
                             float* __restrict__ P, float* __restrict__ Q) {
  const int tid = blockIdx.x * blockDim.x + threadIdx.x;
  if (tid >= NCH * NCHUNK) return;
  const int chunk = tid % NCHUNK;
  const int ch = tid / NCHUNK;
  const int kd = ch % (KD * DI); // k*DI + dch
  const int bk = ch / DI;        // b*KD + k
  float A[NS];
#pragma unroll
  for (int n = 0; n < NS; ++n) A[n] = -__expf(alog[(size_t)kd * NS + n]);
  const float bias = dtb[kd];
  float p[NS], q[NS];
#pragma unroll
  for (int n = 0; n < NS; ++n) { p[n] = 1.f; q[n] = 0.f; }
  const float* u = xs + (size_t)ch * LL + chunk * CLEN;
  const float* dp = dt + (size_t)ch * LL + chunk * CLEN;
  const float* Bp = dbl + ((size_t)bk * CC + RK) * LL + chunk * CLEN;
  for (int l = 0; l < CLEN; ++l) {
    const float dl = softplus_f(dp[l] + bias);
    const float du = dl * u[l];
#pragma unroll
    for (int n = 0; n < NS; ++n) {
      const float a = __expf(dl * A[n]);
      p[n] *= a;
      q[n] = a * q[n] + du * Bp[(size_t)n * LL + l];
    }
  }
  const size_t o = (size_t)tid * NS;
#pragma unroll
  for (int n = 0; n < NS; ++n) { P[o + n] = p[n]; Q[o + n] = q[n]; }
}

__global__ void k_scan_pass2(const float* __restrict__ P, const float* __restrict__ Q,
                             float* __restrict__ h0) {
  const int tid = blockIdx.x * blockDim.x + threadIdx.x;
  if (tid >= NCH * NS) return;
  const int n = tid % NS;
  const int ch = tid / NS;
  float h = 0.f;
  for (int c = 0; c < NCHUNK; ++c) {
    const size_t idx = ((size_t)ch * NCHUNK + c) * NS + n;
    h0[idx] = h;
    h = P[idx] * h + Q[idx];
  }
}

__global__ void k_scan_pass3(float* __restrict__ xs /* u in, y out (aliased) */,
                             const float* __restrict__ dbl, const float* __restrict__ dt,
                             const float* __restrict__ alog,
                             const float* __restrict__ dtb,
                             const float* __restrict__ Ds,
                             const float* __restrict__ h0) {
  const int tid = blockIdx.x * blockDim.x + threadIdx.x;
  if (tid >= NCH * NCHUNK) return;
  const int chunk = tid % NCHUNK;
  const int ch = tid / NCHUNK;
  const int kd = ch % (KD * DI);
  const int bk = ch / DI;
  float A[NS];
#pragma unroll
  for (int n = 0; n < NS; ++n) A[n] = -__expf(alog[(size_t)kd * NS + n]);
  const float bias = dtb[kd];
  const float Dv = Ds[kd];
  float h[NS];
  const size_t ho = ((size_t)ch * NCHUNK + chunk) * NS;
#pragma unroll
  for (int n = 0; n < NS; ++n) h[n] = h0[ho + n];
  float* u = xs + (size_t)ch * LL + chunk * CLEN;
  const float* dp = dt + (size_t)ch * LL + chunk * CLEN;
  const float* Bp = dbl + ((size_t)bk * CC + RK) * LL + chunk * CLEN;
  const float* Cp = dbl + ((size_t)bk * CC + RK + NS) * LL + chunk * CLEN;
  for (int l = 0; l < CLEN; ++l) {
    const float dl = softplus_f(dp[l] + bias);
    const float uu = u[l];
    const float du = dl * uu;
    float y = 0.f;
#pragma unroll
    for (int n = 0; n < NS; ++n) {
      const float a = __expf(dl * A[n]);
      h[n] = a * h[n] + du * Bp[(size_t)n * LL + l];
      y += h[n] * Cp[(size_t)n * LL + l];
    }
    u[l] = y + Dv * uu; // y overwrites u at the same slot (read-then-write)
  }
}

// ---------------------------------------------------------------------------
// K6: merge 4 directions + LayerNorm(192) + gate by z  -> yo (B,L,DI)
// ---------------------------------------------------------------------------
__global__ void k_merge_ln(const float* __restrict__ y, const float* __restrict__ z,
                           const float* __restrict__ g, const float* __restrict__ be,
                           float* __restrict__ yo) {
  __shared__ float s1[256];
  __shared__ float s2[256];
  const int bl = blockIdx.x;
  const int b = bl / LL, l = bl % LL;
  const int c = threadIdx.x;
  const int h = l / WW, w = l % WW;
  const int lwh = w * HH + h;
  float v = 0.f;
  if (c < DI) {
    const size_t base = (size_t)b * KD * DI * LL + (size_t)c * LL;
    const size_t kstr = (size_t)DI * LL;
    v = y[base + l]
      + y[base + 2 * kstr + (LL - 1 - l)]
      + y[base + 1 * kstr + lwh]
      + y[base + 3 * kstr + (LL - 1 - lwh)];
  }
  s1[c] = v;
  s2[c] = v * v;
  __syncthreads();
  for (int s = 128; s >= 1; s >>= 1) {
    if (c < s) { s1[c] += s1[c + s]; s2[c] += s2[c + s]; }
    __syncthreads();
  }
  const float mu = s1[0] * (1.f / DI);
  const float var = s2[0] * (1.f / DI) - mu * mu;
  if (c < DI) {
    float o = (v - mu) * __builtin_amdgcn_rsqf(var + 1e-5f) * g[c] + be[c];
    o *= z[((size_t)b * LL + l) * DI + c];
    yo[((size_t)b * LL + l) * DI + c] = o;
  }
}

// ---------------------------------------------------------------------------
// K7: out = yo (8192x192) @ out_proj_w^T (192x96)
// ---------------------------------------------------------------------------
__global__ void k_outproj(const float* __restrict__ yo, const float* __restrict__ w,
                          float* __restrict__ out) {
  const int wid = (blockIdx.x * blockDim.x + threadIdx.x) >> 5;
  const int lane = threadIdx.x & 31;
  const int TN = DM / 16; // 6
  const int tm = wid / TN, tn = wid % TN;
  if (tm >= (BB * LL) / 16) return;
  const int m0 = tm * 16, n0 = tn * 16;
  const int r = lane & 15, hk = lane >> 4;
  const float* arow = yo + (size_t)(m0 + r) * DI;
  const float* brow = w + (size_t)(n0 + r) * DI;
  v8f acc = {0.f, 0.f, 0.f, 0.f, 0.f, 0.f, 0.f, 0.f};
#pragma unroll 4
  for (int kk = 0; kk < DI; kk += 4) {
    const int kb = kk + 2 * hk;
    v2f a, b;
    a.x = arow[kb];     a.y = arow[kb + 1];
    b.x = brow[kb];     b.y = brow[kb + 1];
    acc = wmma_f32_k4(a, b, acc);
  }
#pragma unroll
  for (int i = 0; i < 8; ++i) {
    const int m = m0 + i + 8 * hk;
    out[(size_t)m * DM + n0 + r] = acc[i];
  }
}

// ---------------------------------------------------------------------------
extern "C" void kernel_launch(void* const* d_in, const int* in_sizes, int n_in,
                              void* d_out, int out_size, void* d_ws, size_t ws_size,
                              hipStream_t stream) {
  (void)in_sizes; (void)n_in; (void)out_size; (void)ws_size;
  const float* x    = (const float*)d_in[0];
  const float* ipw  = (const float*)d_in[1];
  const float* cw   = (const float*)d_in[2];
  const float* cb   = (const float*)d_in[3];
  const float* xpw  = (const float*)d_in[4];
  const float* dtw  = (const float*)d_in[5];
  const float* dtb  = (const float*)d_in[6];
  const float* alog = (const float*)d_in[7];
  const float* Ds   = (const float*)d_in[8];
  const float* ng   = (const float*)d_in[9];
  const float* nb   = (const float*)d_in[10];
  const float* opw  = (const float*)d_in[11];
  float* out = (float*)d_out;

  float* ws = (float*)d_ws;
  size_t o = 0;
  float* xi  = ws + o; o += (size_t)BB * LL * DI;        // conv input; reused as yo
  float* z   = ws + o; o += (size_t)BB * LL * DI;        // silu gate
  float* xs  = ws + o; o += (size_t)BB * KD * DI * LL;   // 4-dir seqs, becomes y
  float* dbl = ws + o; o += (size_t)BB * KD * CC * LL;   // x_dbl (dts|B|C rows)
  float* dt  = ws + o; o += (size_t)BB * KD * DI * LL;   // dt projection
  float* P   = ws + o; o += (size_t)NCH * NCHUNK * NS;   // chunk products
  float* Q   = ws + o; o += (size_t)NCH * NCHUNK * NS;   // chunk offsets
  float* h0  = ws + o; o += (size_t)NCH * NCHUNK * NS;   // chunk initial states
  float* yo  = xi;                                        // xi is dead after conv

  // K1: in_proj GEMM (WMMA): 512 M-tiles x 24 N-tiles, 4 waves/block
  k_inproj<<<((BB * LL) / 16) * ((2 * DI) / 16) / 4, 128, 0, stream>>>(x, ipw, xi, z);
  // K2: depthwise conv + silu + 4-direction scatter
  k_conv_silu_scatter<<<(BB * DI * LL + 255) / 256, 256, 0, stream>>>(xi, cw, cb, xs);
  // K3: x_proj GEMM (WMMA): 8 bk x 3 M-tiles x 256 N-tiles
  k_xdbl<<<(BB * KD * 3 * (LL / 16)) / 4, 128, 0, stream>>>(xs, xpw, dbl);
  // K4: dt_proj GEMM (WMMA): 8 bk x 12 M-tiles x 256 N-tiles
  k_dtproj<<<(BB * KD * (DI / 16) * (LL / 16)) / 4, 128, 0, stream>>>(dbl, dtw, dt);
  // K5: chunked selective scan (3 passes)
  k_scan_pass1<<<(NCH * NCHUNK + 255) / 256, 256, 0, stream>>>(xs, dbl, dt, alog, dtb, P, Q);
  k_scan_pass2<<<(NCH * NS + 255) / 256, 256, 0, stream>>>(P, Q, h0);
  k_scan_pass3<<<(NCH * NCHUNK + 255) / 256, 256, 0, stream>>>(xs, dbl, dt, alog, dtb, Ds, h0);
  // K6: merge 4 directions + LayerNorm + gate
  k_merge_ln<<<BB * LL, 256, 0, stream>>>(xs, z, ng, nb, yo);
  // K7: out_proj GEMM (WMMA): 512 M-tiles x 6 N-tiles
  k_outproj<<<((BB * LL) / 16) * (DM / 16) / 4, 128, 0, stream>>>(yo, opw, out);
}